// TemplateEncoder_54614804136198
// MI455X (gfx1250) — compile-verified
//
#include <hip/hip_runtime.h>
#include <hip/hip_bf16.h>
#include <math.h>

// ---------------- problem constants (from reference) ----------------
constexpr int CB = 2;     // batch
constexpr int CN = 48;    // agents
constexpr int CT = 24;    // timesteps
constexpr int CM = 192;   // map polylines
constexpr int CL = 20;    // points per polyline
constexpr int CH = 128;   // hidden
constexpr int BNT = CB * CN * CT;          // 2304 agent-time rows
constexpr float RADIUS2 = 30.0f * 30.0f;
constexpr float INV_SQRT_H = 0.08838834764831845f; // 1/sqrt(128)

// ---------------- WMMA fp32 16x16x4 helper ----------------
typedef __attribute__((ext_vector_type(2))) float v2f;
typedef __attribute__((ext_vector_type(8))) float v8f;

__device__ __forceinline__ v8f wmma4(v2f a, v2f b, v8f c) {
  // 8 args: (neg_a, A, neg_b, B, c_mod, C, reuse_a, reuse_b)
  return __builtin_amdgcn_wmma_f32_16x16x4_f32(false, a, false, b, (short)0, c,
                                               false, false);
}

__device__ __forceinline__ float relu(float x) { return x > 0.f ? x : 0.f; }

// =====================================================================
// Kernel A: map encoder. One workgroup (128 thr = 4 waves) per (b,m).
//   pts -> relu(2->128) -> relu(128->128 via WMMA) -> max over L
//        -> + embeddings -> relu(128->128) -> (128->128) * mask
// =====================================================================
__global__ void map_encode_kernel(
    const float* __restrict__ polys, const float* __restrict__ pmask,
    const int* __restrict__ ptype, const int* __restrict__ ptl,
    const int* __restrict__ proute,
    const float* __restrict__ mp_w0, const float* __restrict__ mp_b0,
    const float* __restrict__ mp_w1, const float* __restrict__ mp_b1,
    const float* __restrict__ type_emb, const float* __restrict__ tl_emb,
    const float* __restrict__ route_emb,
    const float* __restrict__ mo_w0, const float* __restrict__ mo_b0,
    const float* __restrict__ mo_w1, const float* __restrict__ mo_b1,
    float* __restrict__ map_node, float* __restrict__ map_center) {
  __shared__ float pts[CL * 2];
  __shared__ float h0[32 * CH];
  __shared__ float h1[32 * CH];
  __shared__ float hred[CH];
  __shared__ float g[CH];

  const int bm = blockIdx.x;          // b*CM + m
  const int tid = threadIdx.x;        // 0..127

  if (tid < CL * 2) pts[tid] = polys[bm * CL * 2 + tid];
  __syncthreads();

  if (tid < 2) {  // polyline center = mean over L
    float s = 0.f;
    for (int l = 0; l < CL; ++l) s += pts[l * 2 + tid];
    map_center[bm * 2 + tid] = s / (float)CL;
  }

  // h0 = relu(pts @ mp_w0 + b0), rows 20..31 zero-padded for WMMA
  for (int idx = tid; idx < 32 * CH; idx += 128) {
    const int l = idx >> 7, c = idx & (CH - 1);
    float v = 0.f;
    if (l < CL)
      v = relu(pts[l * 2 + 0] * mp_w0[c] + pts[l * 2 + 1] * mp_w0[CH + c] +
               mp_b0[c]);
    h0[idx] = v;
  }
  __syncthreads();

  // h1 = relu(h0 @ mp_w1 + b1) : 32x128x128 via WMMA, 2x8 tiles over 4 waves
  const int wave = tid >> 5, lane = tid & 31;
  const int half = lane >> 4, id16 = lane & 15;
  for (int tt = wave; tt < 16; tt += 4) {
    const int r0 = (tt & 1) * 16, c0 = (tt >> 1) * 16;
    v8f acc = {};
#pragma unroll 8
    for (int k0 = 0; k0 < CH; k0 += 4) {
      v2f a, b;
      a.x = h0[(r0 + id16) * CH + k0 + 2 * half];
      a.y = h0[(r0 + id16) * CH + k0 + 2 * half + 1];
      b.x = mp_w1[(k0 + 2 * half) * CH + c0 + id16];
      b.y = mp_w1[(k0 + 2 * half + 1) * CH + c0 + id16];
      acc = wmma4(a, b, acc);
    }
#pragma unroll
    for (int r = 0; r < 8; ++r) {
      const int row = r0 + r + 8 * half, col = c0 + id16;
      h1[row * CH + col] = relu(acc[r] + mp_b1[col]);
    }
  }
  __syncthreads();

  // max-pool over the 20 valid rows, then add categorical embeddings
  {
    float mx = h1[tid];
    for (int l = 1; l < CL; ++l) mx = fmaxf(mx, h1[l * CH + tid]);
    int pt = ptype[bm];  pt = pt < 0 ? 0 : (pt > 3 ? 3 : pt);
    int tl = ptl[bm];    tl = tl < 0 ? 0 : (tl > 7 ? 7 : tl);
    int rt = proute[bm]; rt = rt < 0 ? 0 : (rt > 1 ? 1 : rt);
    hred[tid] = mx + type_emb[pt * CH + tid] + tl_emb[tl * CH + tid] +
                route_emb[rt * CH + tid];
  }
  __syncthreads();

  {  // g = relu(hred @ mo_w0 + b0)   (vector-matrix: VALU is right-sized)
    float s = mo_b0[tid];
#pragma unroll 4
    for (int k = 0; k < CH; ++k) s += hred[k] * mo_w0[k * CH + tid];
    g[tid] = relu(s);
  }
  __syncthreads();

  {  // map_node = (g @ mo_w1 + b1) * mask
    float s = mo_b1[tid];
#pragma unroll 4
    for (int k = 0; k < CH; ++k) s += g[k] * mo_w1[k * CH + tid];
    const float msk = pmask[bm] > 0.5f ? 1.f : 0.f;
    map_node[bm * CH + tid] = s * msk;
  }
}

// =====================================================================
// Kernel B: agent encoder. 32 rows per workgroup (72 blocks x 128 thr).
//   a0 = relu(5->128) ; a1 = relu(a0@W1+b1) WMMA ; a_emb = (a1@W2+b2)*mask
// =====================================================================
__global__ void agent_encode_kernel(
    const float* __restrict__ astate, const float* __restrict__ amask,
    const float* __restrict__ ae_w0, const float* __restrict__ ae_b0,
    const float* __restrict__ ae_w1, const float* __restrict__ ae_b1,
    const float* __restrict__ ae_w2, const float* __restrict__ ae_b2,
    float* __restrict__ a_emb) {
  __shared__ float feat[32 * 5];
  __shared__ float a0[32 * CH];
  __shared__ float a1[32 * CH];
  const int row0 = blockIdx.x * 32;
  const int tid = threadIdx.x;
  if (tid < 32 * 5) feat[tid] = astate[row0 * 5 + tid];
  __syncthreads();

  for (int idx = tid; idx < 32 * CH; idx += 128) {
    const int r = idx >> 7, c = idx & (CH - 1);
    float v = ae_b0[c];
#pragma unroll
    for (int i = 0; i < 5; ++i) v += feat[r * 5 + i] * ae_w0[i * CH + c];
    a0[idx] = relu(v);
  }
  __syncthreads();

  const int wave = tid >> 5, lane = tid & 31;
  const int half = lane >> 4, id16 = lane & 15;
  for (int tt = wave; tt < 16; tt += 4) {
    const int r0 = (tt & 1) * 16, c0 = (tt >> 1) * 16;
    v8f acc = {};
#pragma unroll 8
    for (int k0 = 0; k0 < CH; k0 += 4) {
      v2f a, b;
      a.x = a0[(r0 + id16) * CH + k0 + 2 * half];
      a.y = a0[(r0 + id16) * CH + k0 + 2 * half + 1];
      b.x = ae_w1[(k0 + 2 * half) * CH + c0 + id16];
      b.y = ae_w1[(k0 + 2 * half + 1) * CH + c0 + id16];
      acc = wmma4(a, b, acc);
    }
#pragma unroll
    for (int r = 0; r < 8; ++r) {
      const int row = r0 + r + 8 * half, col = c0 + id16;
      a1[row * CH + col] = relu(acc[r] + ae_b1[col]);
    }
  }
  __syncthreads();

  for (int tt = wave; tt < 16; tt += 4) {
    const int r0 = (tt & 1) * 16, c0 = (tt >> 1) * 16;
    v8f acc = {};
#pragma unroll 8
    for (int k0 = 0; k0 < CH; k0 += 4) {
      v2f a, b;
      a.x = a1[(r0 + id16) * CH + k0 + 2 * half];
      a.y = a1[(r0 + id16) * CH + k0 + 2 * half + 1];
      b.x = ae_w2[(k0 + 2 * half) * CH + c0 + id16];
      b.y = ae_w2[(k0 + 2 * half + 1) * CH + c0 + id16];
      acc = wmma4(a, b, acc);
    }
#pragma unroll
    for (int r = 0; r < 8; ++r) {
      const int row = r0 + r + 8 * half, col = c0 + id16;
      const int grow = row0 + row;
      const float msk = amask[grow] > 0.5f ? 1.f : 0.f;
      a_emb[grow * CH + col] = (acc[r] + ae_b2[col]) * msk;
    }
  }
}

// =====================================================================
// Kernel C: per-(bnt,m) pairwise map score MLP (3->128->1). K too skinny
// for WMMA; weights are wave-uniform (scalar loads).
// =====================================================================
__global__ void map_rel_score_kernel(
    const float* __restrict__ astate, const float* __restrict__ center,
    const float* __restrict__ mr_w0, const float* __restrict__ mr_b0,
    const float* __restrict__ mr_w1, const float* __restrict__ mr_b1,
    float* __restrict__ logits) {
  const int i = blockIdx.x * blockDim.x + threadIdx.x;
  if (i >= BNT * CM) return;
  const int bnt = i / CM, m = i % CM;
  const int b = bnt / (CN * CT);
  const float px = astate[bnt * 5 + 0], py = astate[bnt * 5 + 1];
  const float rx = center[(b * CM + m) * 2 + 0] - px;
  const float ry = center[(b * CM + m) * 2 + 1] - py;
  const float d = sqrtf(rx * rx + ry * ry);
  float s = mr_b1[0];
#pragma unroll 4
  for (int k = 0; k < CH; ++k) {
    const float h = relu(rx * mr_w0[k] + ry * mr_w0[CH + k] +
                         d * mr_w0[2 * CH + k] + mr_b0[k]);
    s += h * mr_w1[k];
  }
  logits[i] = s;
}

// =====================================================================
// Kernel D: logits += (a_emb . map_node)/sqrt(H).  WMMA GEMM 2304x192,K=128.
// One 16x16 tile per wave; 432 blocks x 4 waves = 1728 tiles exactly.
// =====================================================================
__global__ void map_dot_kernel(const float* __restrict__ a_emb,
                               const float* __restrict__ map_node,
                               float* __restrict__ logits) {
  const int tid = threadIdx.x, wave = tid >> 5, lane = tid & 31;
  const int half = lane >> 4, id16 = lane & 15;
  const int tile = blockIdx.x * 4 + wave;       // 0..1727
  const int row0 = (tile / 12) * 16, c0 = (tile % 12) * 16;
  const int b = row0 / (CN * CT);
  v8f acc = {};
#pragma unroll 8
  for (int k0 = 0; k0 < CH; k0 += 4) {
    v2f a, bb;
    a.x = a_emb[(row0 + id16) * CH + k0 + 2 * half];
    a.y = a_emb[(row0 + id16) * CH + k0 + 2 * half + 1];
    // B = map_node^T : B[k][n] = map_node[b, c0+n, k]
    bb.x = map_node[(b * CM + c0 + id16) * CH + k0 + 2 * half];
    bb.y = map_node[(b * CM + c0 + id16) * CH + k0 + 2 * half + 1];
    acc = wmma4(a, bb, acc);
  }
#pragma unroll
  for (int r = 0; r < 8; ++r) {
    const int row = row0 + r + 8 * half, col = c0 + id16;
    logits[row * CM + col] += acc[r] * INV_SQRT_H;
  }
}

// =====================================================================
// Kernel E: masked softmax over M=192, one wave32 per row (6 elems/lane).
// =====================================================================
__global__ void map_softmax_kernel(const float* __restrict__ pmask,
                                   float* __restrict__ logits) {
  const int wave = threadIdx.x >> 5, lane = threadIdx.x & 31;
  const int row = blockIdx.x * 8 + wave;        // 0..2303
  const int b = row / (CN * CT);
  float v[6];
  float mx = -3.402823466e38f;
#pragma unroll
  for (int j = 0; j < 6; ++j) {
    const int m = j * 32 + lane;
    const bool mk = pmask[b * CM + m] > 0.5f;
    v[j] = mk ? logits[row * CM + m] : -1.0e9f;
    mx = fmaxf(mx, v[j]);
  }
  for (int off = 16; off > 0; off >>= 1) mx = fmaxf(mx, __shfl_xor(mx, off, 32));
  float sum = 0.f;
#pragma unroll
  for (int j = 0; j < 6; ++j) { v[j] = __expf(v[j] - mx); sum += v[j]; }
  for (int off = 16; off > 0; off >>= 1) sum += __shfl_xor(sum, off, 32);
  const float isum = 1.f / sum;
#pragma unroll
  for (int j = 0; j < 6; ++j) logits[row * CM + j * 32 + lane] = v[j] * isum;
}

// =====================================================================
// Kernel F: map_ctx = attn(2304x192) @ map_node_b(192x128). WMMA, K=192.
// 288 blocks x 4 waves = 1152 tiles (144 row x 8 col).
// =====================================================================
__global__ void map_ctx_kernel(const float* __restrict__ attn,
                               const float* __restrict__ map_node,
                               float* __restrict__ out_ctx) {
  const int tid = threadIdx.x, wave = tid >> 5, lane = tid & 31;
  const int half = lane >> 4, id16 = lane & 15;
  const int tile = blockIdx.x * 4 + wave;       // 0..1151
  const int row0 = (tile / 8) * 16, c0 = (tile % 8) * 16;
  const int b = row0 / (CN * CT);
  v8f acc = {};
#pragma unroll 8
  for (int k0 = 0; k0 < CM; k0 += 4) {
    v2f a, bb;
    a.x = attn[(row0 + id16) * CM + k0 + 2 * half];
    a.y = attn[(row0 + id16) * CM + k0 + 2 * half + 1];
    bb.x = map_node[(b * CM + k0 + 2 * half) * CH + c0 + id16];
    bb.y = map_node[(b * CM + k0 + 2 * half + 1) * CH + c0 + id16];
    acc = wmma4(a, bb, acc);
  }
#pragma unroll
  for (int r = 0; r < 8; ++r) {
    const int row = row0 + r + 8 * half, col = c0 + id16;
    out_ctx[row * CH + col] = acc[r];
  }
}

// =====================================================================
// Kernel G: neighbor pairwise score MLP (5->128->1) per (bnt, j).
// =====================================================================
__global__ void nbr_score_kernel(const float* __restrict__ astate,
                                 const float* __restrict__ nr_w0,
                                 const float* __restrict__ nr_b0,
                                 const float* __restrict__ nr_w1,
                                 const float* __restrict__ nr_b1,
                                 float* __restrict__ logits) {
  const int i = blockIdx.x * blockDim.x + threadIdx.x;
  if (i >= BNT * CN) return;
  const int bnt = i / CN, j = i % CN;
  const int b = bnt / (CN * CT);
  const int t = bnt % CT;
  const int bjt = (b * CN + j) * CT + t;
  const float dx = astate[bnt * 5 + 0] - astate[bjt * 5 + 0];
  const float dy = astate[bnt * 5 + 1] - astate[bjt * 5 + 1];
  const float dvx = astate[bnt * 5 + 3] - astate[bjt * 5 + 3];
  const float dvy = astate[bnt * 5 + 4] - astate[bjt * 5 + 4];
  const float d = sqrtf(dx * dx + dy * dy);
  float s = nr_b1[0];
#pragma unroll 4
  for (int k = 0; k < CH; ++k) {
    const float h = relu(dx * nr_w0[k] + dy * nr_w0[CH + k] +
                         dvx * nr_w0[2 * CH + k] + dvy * nr_w0[3 * CH + k] +
                         d * nr_w0[4 * CH + k] + nr_b0[k]);
    s += h * nr_w1[k];
  }
  logits[i] = s;
}

// =====================================================================
// Kernel H: logits_nbr += (a_emb_i . a_emb_j)/sqrt(H) per (b,t).
// Per (b,t): 48x48, K=128 -> 3x3 tiles. 108 blocks x 4 waves = 432 tiles.
// =====================================================================
__global__ void nbr_dot_kernel(const float* __restrict__ a_emb,
                               float* __restrict__ logits) {
  const int tid = threadIdx.x, wave = tid >> 5, lane = tid & 31;
  const int half = lane >> 4, id16 = lane & 15;
  const int tile = blockIdx.x * 4 + wave;       // 0..431
  const int bt = tile / 9, sub = tile % 9;
  const int b = bt / CT, t = bt % CT;
  const int n0 = (sub / 3) * 16, j0 = (sub % 3) * 16;
  v8f acc = {};
#pragma unroll 8
  for (int k0 = 0; k0 < CH; k0 += 4) {
    v2f a, bb;
    a.x = a_emb[((b * CN + n0 + id16) * CT + t) * CH + k0 + 2 * half];
    a.y = a_emb[((b * CN + n0 + id16) * CT + t) * CH + k0 + 2 * half + 1];
    bb.x = a_emb[((b * CN + j0 + id16) * CT + t) * CH + k0 + 2 * half];
    bb.y = a_emb[((b * CN + j0 + id16) * CT + t) * CH + k0 + 2 * half + 1];
    acc = wmma4(a, bb, acc);
  }
#pragma unroll
  for (int r = 0; r < 8; ++r) {
    const int n = n0 + r + 8 * half, j = j0 + id16;
    logits[((b * CN + n) * CT + t) * CN + j] += acc[r] * INV_SQRT_H;
  }
}

// =====================================================================
// Kernel I: neighbor masked softmax over j (48), one wave32 per (bnt) row.
// Mask recomputed inline: valid_i & valid_j & (dist<=R) & (j != n).
// =====================================================================
__global__ void nbr_softmax_kernel(const float* __restrict__ astate,
                                   const float* __restrict__ amask,
                                   float* __restrict__ logits) {
  const int wave = threadIdx.x >> 5, lane = threadIdx.x & 31;
  const int row = blockIdx.x * 8 + wave;        // bnt, 0..2303
  const int b = row / (CN * CT);
  const int rem = row % (CN * CT);
  const int n = rem / CT, t = rem % CT;
  const float px = astate[row * 5 + 0], py = astate[row * 5 + 1];
  const bool vi = amask[row] > 0.5f;

  float v[2];
#pragma unroll
  for (int p = 0; p < 2; ++p) {
    const int j = p * 32 + lane;
    if (j < CN) {
      const int bjt = (b * CN + j) * CT + t;
      const float dx = px - astate[bjt * 5 + 0];
      const float dy = py - astate[bjt * 5 + 1];
      const bool mk = vi && (amask[bjt] > 0.5f) &&
                      (dx * dx + dy * dy <= RADIUS2) && (j != n);
      v[p] = mk ? logits[row * CN + j] : -1.0e9f;
    } else {
      v[p] = -1.0e30f;  // exp underflows to 0; never written back
    }
  }
  float mx = fmaxf(v[0], v[1]);
  for (int off = 16; off > 0; off >>= 1) mx = fmaxf(mx, __shfl_xor(mx, off, 32));
  float sum = 0.f;
#pragma unroll
  for (int p = 0; p < 2; ++p) { v[p] = __expf(v[p] - mx); sum += v[p]; }
  for (int off = 16; off > 0; off >>= 1) sum += __shfl_xor(sum, off, 32);
  const float isum = 1.f / sum;
  logits[row * CN + lane] = v[0] * isum;
  if (lane < CN - 32) logits[row * CN + 32 + lane] = v[1] * isum;
}

// =====================================================================
// Kernel J: nbr_ctx = W(48x48) @ a_emb_t(48x128) per (b,t). WMMA K=48.
// 288 blocks x 4 waves = 1152 tiles (48 bt x 3 row x 8 col).
// =====================================================================
__global__ void nbr_ctx_kernel(const float* __restrict__ attn,
                               const float* __restrict__ a_emb,
                               float* __restrict__ out_ctx) {
  const int tid = threadIdx.x, wave = tid >> 5, lane = tid & 31;
  const int half = lane >> 4, id16 = lane & 15;
  const int tile = blockIdx.x * 4 + wave;       // 0..1151
  const int bt = tile / 24, sub = tile % 24;
  const int b = bt / CT, t = bt % CT;
  const int n0 = (sub / 8) * 16, c0 = (sub % 8) * 16;
  v8f acc = {};
#pragma unroll 4
  for (int k0 = 0; k0 < CN; k0 += 4) {
    v2f a, bb;
    a.x = attn[((b * CN + n0 + id16) * CT + t) * CN + k0 + 2 * half];
    a.y = attn[((b * CN + n0 + id16) * CT + t) * CN + k0 + 2 * half + 1];
    bb.x = a_emb[((b * CN + k0 + 2 * half) * CT + t) * CH + c0 + id16];
    bb.y = a_emb[((b * CN + k0 + 2 * half + 1) * CT + t) * CH + c0 + id16];
    acc = wmma4(a, bb, acc);
  }
#pragma unroll
  for (int r = 0; r < 8; ++r) {
    const int n = n0 + r + 8 * half;
    out_ctx[((b * CN + n) * CT + t) * CH + c0 + id16] = acc[r];
  }
}

// =====================================================================
// Kernel K: tau hidden = relu(concat(a_emb,map_ctx,nbr_ctx) @ to_w0 + b0).
// WMMA, K=384 gathered from 3 sources (each k-step lies in one source).
// 288 blocks x 4 waves = 1152 tiles (144 row x 8 col).
// =====================================================================
__global__ void tau_hidden_kernel(const float* __restrict__ a_emb,
                                  const float* __restrict__ mctx,
                                  const float* __restrict__ nctx,
                                  const float* __restrict__ to_w0,
                                  const float* __restrict__ to_b0,
                                  float* __restrict__ hidden) {
  const int tid = threadIdx.x, wave = tid >> 5, lane = tid & 31;
  const int half = lane >> 4, id16 = lane & 15;
  const int tile = blockIdx.x * 4 + wave;       // 0..1151
  const int row0 = (tile / 8) * 16, c0 = (tile % 8) * 16;
  v8f acc = {};
#pragma unroll 8
  for (int k0 = 0; k0 < 3 * CH; k0 += 4) {
    const int seg = k0 >> 7, kk = (k0 & (CH - 1)) + 2 * half;
    const float* S = (seg == 0) ? a_emb : (seg == 1) ? mctx : nctx;
    v2f a, bb;
    a.x = S[(row0 + id16) * CH + kk];
    a.y = S[(row0 + id16) * CH + kk + 1];
    bb.x = to_w0[(k0 + 2 * half) * CH + c0 + id16];
    bb.y = to_w0[(k0 + 2 * half + 1) * CH + c0 + id16];
    acc = wmma4(a, bb, acc);
  }
#pragma unroll
  for (int r = 0; r < 8; ++r) {
    const int row = row0 + r + 8 * half, col = c0 + id16;
    hidden[row * CH + col] = relu(acc[r] + to_b0[col]);
  }
}

// =====================================================================
// Kernel L: tau = (hidden @ to_w1 + b1) * mask. WMMA 2304x64, K=128.
// 144 blocks x 4 waves = 576 tiles (144 row x 4 col).
// =====================================================================
__global__ void tau_out_kernel(const float* __restrict__ hidden,
                               const float* __restrict__ amask,
                               const float* __restrict__ to_w1,
                               const float* __restrict__ to_b1,
                               float* __restrict__ tau) {
  const int tid = threadIdx.x, wave = tid >> 5, lane = tid & 31;
  const int half = lane >> 4, id16 = lane & 15;
  const int tile = blockIdx.x * 4 + wave;       // 0..575
  const int row0 = (tile / 4) * 16, c0 = (tile % 4) * 16;
  v8f acc = {};
#pragma unroll 8
  for (int k0 = 0; k0 < CH; k0 += 4) {
    v2f a, bb;
    a.x = hidden[(row0 + id16) * CH + k0 + 2 * half];
    a.y = hidden[(row0 + id16) * CH + k0 + 2 * half + 1];
    bb.x = to_w1[(k0 + 2 * half) * 64 + c0 + id16];
    bb.y = to_w1[(k0 + 2 * half + 1) * 64 + c0 + id16];
    acc = wmma4(a, bb, acc);
  }
#pragma unroll
  for (int r = 0; r < 8; ++r) {
    const int row = row0 + r + 8 * half, col = c0 + id16;
    const float msk = amask[row] > 0.5f ? 1.f : 0.f;
    tau[row * 64 + col] = (acc[r] + to_b1[col]) * msk;
  }
}

// =====================================================================
extern "C" void kernel_launch(void* const* d_in, const int* in_sizes, int n_in,
                              void* d_out, int out_size, void* d_ws,
                              size_t ws_size, hipStream_t stream) {
  // inputs in setup_inputs() order
  const float* agents_state  = (const float*)d_in[0];
  const float* agents_mask   = (const float*)d_in[1];
  const float* map_polylines = (const float*)d_in[2];
  const float* map_poly_mask = (const float*)d_in[3];
  const int*   map_poly_type = (const int*)d_in[4];
  const int*   map_tl_status = (const int*)d_in[5];
  const int*   map_on_route  = (const int*)d_in[6];
  const float* ae_w0 = (const float*)d_in[7];
  const float* ae_b0 = (const float*)d_in[8];
  const float* ae_w1 = (const float*)d_in[9];
  const float* ae_b1 = (const float*)d_in[10];
  const float* ae_w2 = (const float*)d_in[11];
  const float* ae_b2 = (const float*)d_in[12];
  const float* mp_w0 = (const float*)d_in[13];
  const float* mp_b0 = (const float*)d_in[14];
  const float* mp_w1 = (const float*)d_in[15];
  const float* mp_b1 = (const float*)d_in[16];
  const float* type_emb  = (const float*)d_in[17];
  const float* tl_emb    = (const float*)d_in[18];
  const float* route_emb = (const float*)d_in[19];
  const float* mo_w0 = (const float*)d_in[20];
  const float* mo_b0 = (const float*)d_in[21];
  const float* mo_w1 = (const float*)d_in[22];
  const float* mo_b1 = (const float*)d_in[23];
  const float* mr_w0 = (const float*)d_in[24];
  const float* mr_b0 = (const float*)d_in[25];
  const float* mr_w1 = (const float*)d_in[26];
  const float* mr_b1 = (const float*)d_in[27];
  const float* nr_w0 = (const float*)d_in[28];
  const float* nr_b0 = (const float*)d_in[29];
  const float* nr_w1 = (const float*)d_in[30];
  const float* nr_b1 = (const float*)d_in[31];
  const float* to_w0 = (const float*)d_in[32];
  const float* to_b0 = (const float*)d_in[33];
  const float* to_w1 = (const float*)d_in[34];
  const float* to_b1 = (const float*)d_in[35];

  // outputs: tau (2304x64) | map_ctx (2304x128) | nbr_ctx (2304x128)
  float* tau  = (float*)d_out;
  float* mctx = tau + BNT * 64;
  float* nctx = mctx + BNT * CH;

  // workspace layout (floats)
  float* w          = (float*)d_ws;
  float* a_emb      = w;                         // 2304*128 = 294912
  float* map_node   = a_emb + BNT * CH;          // 384*128  =  49152
  float* map_center = map_node + CB * CM * CH;   // 384*2    =    768
  float* lg_map     = map_center + CB * CM * 2;  // 2304*192 = 442368
  float* lg_nbr     = lg_map + BNT * CM;         // 2304*48  = 110592
  float* tau_hid    = lg_nbr + BNT * CN;         // 2304*128 = 294912

  map_encode_kernel<<<CB * CM, 128, 0, stream>>>(
      map_polylines, map_poly_mask, map_poly_type, map_tl_status, map_on_route,
      mp_w0, mp_b0, mp_w1, mp_b1, type_emb, tl_emb, route_emb,
      mo_w0, mo_b0, mo_w1, mo_b1, map_node, map_center);

  agent_encode_kernel<<<BNT / 32, 128, 0, stream>>>(
      agents_state, agents_mask, ae_w0, ae_b0, ae_w1, ae_b1, ae_w2, ae_b2,
      a_emb);

  map_rel_score_kernel<<<(BNT * CM) / 256, 256, 0, stream>>>(
      agents_state, map_center, mr_w0, mr_b0, mr_w1, mr_b1, lg_map);

  map_dot_kernel<<<(BNT / 16) * (CM / 16) / 4, 128, 0, stream>>>(
      a_emb, map_node, lg_map);

  map_softmax_kernel<<<BNT / 8, 256, 0, stream>>>(map_poly_mask, lg_map);

  map_ctx_kernel<<<(BNT / 16) * (CH / 16) / 4, 128, 0, stream>>>(
      lg_map, map_node, mctx);

  nbr_score_kernel<<<(BNT * CN) / 256, 256, 0, stream>>>(
      agents_state, nr_w0, nr_b0, nr_w1, nr_b1, lg_nbr);

  nbr_dot_kernel<<<(CB * CT * 9) / 4, 128, 0, stream>>>(a_emb, lg_nbr);

  nbr_softmax_kernel<<<BNT / 8, 256, 0, stream>>>(agents_state, agents_mask,
                                                  lg_nbr);

  nbr_ctx_kernel<<<(CB * CT * 24) / 4, 128, 0, stream>>>(lg_nbr, a_emb, nctx);

  tau_hidden_kernel<<<(BNT / 16) * (CH / 16) / 4, 128, 0, stream>>>(
      a_emb, mctx, nctx, to_w0, to_b0, tau_hid);

  tau_out_kernel<<<(BNT / 16) * (64 / 16) / 4, 128, 0, stream>>>(
      tau_hid, agents_mask, to_w1, to_b1, tau);
}